// GCN_50835232915957
// MI455X (gfx1250) — compile-verified
//
#include <hip/hip_runtime.h>
#include <hip/hip_bf16.h>
#include <math.h>

typedef __attribute__((ext_vector_type(2))) float v2f;
typedef __attribute__((ext_vector_type(8))) float v8f;

#define NFEAT 128
#define HID   24
#define HPAD  32
#define NCLS  16

// ---------- degree / normalization ----------
__global__ __launch_bounds__(256) void k_init_deg(float* deg, int n) {
  int i = blockIdx.x * blockDim.x + threadIdx.x;
  if (i < n) deg[i] = 1.0f;  // self loop
}

__global__ __launch_bounds__(256) void k_count_deg(const int* __restrict__ dst,
                                                   float* deg, int E) {
  int e = blockIdx.x * blockDim.x + threadIdx.x;
  if (e < E) atomicAdd(&deg[dst[e]], 1.0f);
}

__global__ __launch_bounds__(256) void k_rsqrt_inplace(float* deg, int n) {
  int i = blockIdx.x * blockDim.x + threadIdx.x;
  if (i < n) deg[i] = rsqrtf(deg[i]);  // deg >= 1 always (self loop)
}

// ---------- GEMM1: H1[N,24] = X[N,128] @ W1[128,24] via V_WMMA_F32_16X16X4_F32 ----------
__global__ __launch_bounds__(128) void k_gemm1_wmma(const float* __restrict__ X,
                                                    const float* __restrict__ W1,
                                                    float* __restrict__ H1,
                                                    int nNodes) {
  __shared__ float ws[NFEAT * HPAD];  // W1 padded to 32 cols, 16 KB LDS
  int tid = threadIdx.x;
  for (int i = tid; i < NFEAT * HPAD; i += 128) {
    int r = i >> 5, c = i & 31;
    ws[i] = (c < HID) ? W1[r * HID + c] : 0.0f;
  }
  __syncthreads();

  int wave = tid >> 5, lane = tid & 31;
  int row0 = (blockIdx.x * 4 + wave) * 16;
  if (row0 >= nNodes) return;  // wave-uniform: EXEC stays all-ones for WMMA

  int mi = lane & 15;   // M for A-frag, N for B/C frags
  int kh = lane >> 4;   // K-half selector

  v8f acc0 = {};  // output cols 0..15
  v8f acc1 = {};  // output cols 16..31 (only 16..23 valid)
  const float* xr = X + (size_t)(row0 + mi) * NFEAT;
  for (int k0 = 0; k0 < NFEAT; k0 += 4) {
    int kk = k0 + 2 * kh;
    v2f a = *(const v2f*)(xr + kk);            // A: K=kk, kk+1
    v2f b0, b1;
    b0.x = ws[kk * HPAD + mi];                 // B: K=kk,   N=mi
    b0.y = ws[(kk + 1) * HPAD + mi];           //    K=kk+1, N=mi
    b1.x = ws[kk * HPAD + 16 + mi];
    b1.y = ws[(kk + 1) * HPAD + 16 + mi];
    acc0 = __builtin_amdgcn_wmma_f32_16x16x4_f32(false, a, false, b0, (short)0, acc0, false, false);
    acc1 = __builtin_amdgcn_wmma_f32_16x16x4_f32(false, a, false, b1, (short)0, acc1, false, false);
  }
  // C layout: lane -> N=mi, VGPR r -> M = r + 8*kh
  for (int r = 0; r < 8; ++r) {
    int row = row0 + r + 8 * kh;
    H1[(size_t)row * HID + mi] = acc0[r];
    if (mi < 8) H1[(size_t)row * HID + 16 + mi] = acc1[r];
  }
}

// ---------- self-loop init: agg1 = h1 * dinv^2 ----------
__global__ __launch_bounds__(256) void k_self1(const float* __restrict__ h1,
                                               const float* __restrict__ dinv,
                                               float* __restrict__ agg1, int n) {
  int i = blockIdx.x * blockDim.x + threadIdx.x;
  if (i >= n * HID) return;
  float di = dinv[i / HID];
  agg1[i] = h1[i] * di * di;
}

// ---------- edge scatter layer 1: 8 lanes per edge, 3 cols each ----------
__global__ __launch_bounds__(256) void k_scatter1(const int* __restrict__ dst,
                                                  const int* __restrict__ src,
                                                  const float* __restrict__ dinv,
                                                  const float* __restrict__ h1,
                                                  float* agg1, int E) {
  int gid = blockIdx.x * blockDim.x + threadIdx.x;
  int e = gid >> 3;
  if (e >= E) return;
  int part = gid & 7;
  int s = src[e], d = dst[e];
  float nrm = dinv[s] * dinv[d];
  int c0 = part * 3;
  const float* hs = h1 + (size_t)s * HID + c0;
  float* od = agg1 + (size_t)d * HID + c0;
  atomicAdd(od + 0, hs[0] * nrm);
  atomicAdd(od + 1, hs[1] * nrm);
  atomicAdd(od + 2, hs[2] * nrm);
}

// ---------- bias + ReLU (in place) ----------
__global__ __launch_bounds__(256) void k_bias_relu(float* a, const float* __restrict__ b, int n) {
  int i = blockIdx.x * blockDim.x + threadIdx.x;
  if (i >= n * HID) return;
  a[i] = fmaxf(a[i] + b[i % HID], 0.0f);
}

// ---------- GEMM2: H2[N,16] = A1[N,24] @ W2[24,16] via WMMA ----------
__global__ __launch_bounds__(128) void k_gemm2_wmma(const float* __restrict__ A1,
                                                    const float* __restrict__ W2,
                                                    float* __restrict__ H2,
                                                    int nNodes) {
  __shared__ float ws[HID * NCLS];  // 1.5 KB
  int tid = threadIdx.x;
  for (int i = tid; i < HID * NCLS; i += 128) ws[i] = W2[i];
  __syncthreads();

  int wave = tid >> 5, lane = tid & 31;
  int row0 = (blockIdx.x * 4 + wave) * 16;
  if (row0 >= nNodes) return;

  int mi = lane & 15, kh = lane >> 4;
  v8f acc = {};
  const float* ar = A1 + (size_t)(row0 + mi) * HID;
  for (int k0 = 0; k0 < HID; k0 += 4) {
    int kk = k0 + 2 * kh;
    v2f a = *(const v2f*)(ar + kk);
    v2f b;
    b.x = ws[kk * NCLS + mi];
    b.y = ws[(kk + 1) * NCLS + mi];
    acc = __builtin_amdgcn_wmma_f32_16x16x4_f32(false, a, false, b, (short)0, acc, false, false);
  }
  for (int r = 0; r < 8; ++r)
    H2[(size_t)(row0 + r + 8 * kh) * NCLS + mi] = acc[r];
}

// ---------- self-loop init layer 2 ----------
__global__ __launch_bounds__(256) void k_self2(const float* __restrict__ h2,
                                               const float* __restrict__ dinv,
                                               float* __restrict__ agg2, int n) {
  int i = blockIdx.x * blockDim.x + threadIdx.x;
  if (i >= n * NCLS) return;
  float di = dinv[i / NCLS];
  agg2[i] = h2[i] * di * di;
}

// ---------- edge scatter layer 2: 4 lanes per edge, 4 cols each ----------
__global__ __launch_bounds__(256) void k_scatter2(const int* __restrict__ dst,
                                                  const int* __restrict__ src,
                                                  const float* __restrict__ dinv,
                                                  const float* __restrict__ h2,
                                                  float* agg2, int E) {
  int gid = blockIdx.x * blockDim.x + threadIdx.x;
  int e = gid >> 2;
  if (e >= E) return;
  int part = gid & 3;
  int s = src[e], d = dst[e];
  float nrm = dinv[s] * dinv[d];
  int c0 = part * 4;
  const float* hs = h2 + (size_t)s * NCLS + c0;
  float* od = agg2 + (size_t)d * NCLS + c0;
  atomicAdd(od + 0, hs[0] * nrm);
  atomicAdd(od + 1, hs[1] * nrm);
  atomicAdd(od + 2, hs[2] * nrm);
  atomicAdd(od + 3, hs[3] * nrm);
}

// ---------- bias + log_softmax (one node per thread, 16 classes) ----------
__global__ __launch_bounds__(256) void k_logsoftmax(const float* __restrict__ agg2,
                                                    const float* __restrict__ b2,
                                                    float* __restrict__ out, int n) {
  int node = blockIdx.x * blockDim.x + threadIdx.x;
  if (node >= n) return;
  const float* r = agg2 + (size_t)node * NCLS;
  float v[NCLS];
  float m = -INFINITY;
  for (int c = 0; c < NCLS; ++c) { v[c] = r[c] + b2[c]; m = fmaxf(m, v[c]); }
  float s = 0.0f;
  for (int c = 0; c < NCLS; ++c) s += expf(v[c] - m);
  float l = logf(s);
  float* o = out + (size_t)node * NCLS;
  for (int c = 0; c < NCLS; ++c) o[c] = v[c] - m - l;
}

extern "C" void kernel_launch(void* const* d_in, const int* in_sizes, int n_in,
                              void* d_out, int out_size, void* d_ws, size_t ws_size,
                              hipStream_t stream) {
  (void)n_in; (void)out_size; (void)ws_size;
  const float* X  = (const float*)d_in[0];
  const int*   EI = (const int*)d_in[1];  // [2,E] flat: row(dst) then col(src)
  const float* W1 = (const float*)d_in[2];
  const float* B1 = (const float*)d_in[3];
  const float* W2 = (const float*)d_in[4];
  const float* B2 = (const float*)d_in[5];

  const int nNodes = in_sizes[0] / NFEAT;   // 100000
  const int E = in_sizes[1] / 2;            // 3200000
  const int* dst = EI;
  const int* src = EI + E;

  // workspace layout (256B aligned slices)
  char* ws = (char*)d_ws;
  size_t off = 0;
  auto alloc = [&](size_t bytes) -> void* {
    void* p = ws + off;
    off = (off + bytes + 255) & ~(size_t)255;
    return p;
  };
  float* dinv = (float*)alloc((size_t)nNodes * 4);          // deg -> rsqrt in place
  float* h1   = (float*)alloc((size_t)nNodes * HID * 4);
  float* agg1 = (float*)alloc((size_t)nNodes * HID * 4);    // becomes a1 after bias+relu
  float* h2   = (float*)alloc((size_t)nNodes * NCLS * 4);
  float* agg2 = (float*)alloc((size_t)nNodes * NCLS * 4);

  const int T = 256;
  int nbN   = (nNodes + T - 1) / T;
  int nbE   = (E + T - 1) / T;
  int nbNH  = (nNodes * HID + T - 1) / T;
  int nbNC  = (nNodes * NCLS + T - 1) / T;
  int mtiles = (nNodes + 15) / 16;
  int gemmB  = (mtiles + 3) / 4;  // 4 waves (M-tiles) per 128-thread block

  // normalization
  k_init_deg<<<nbN, T, 0, stream>>>(dinv, nNodes);
  k_count_deg<<<nbE, T, 0, stream>>>(dst, dinv, E);
  k_rsqrt_inplace<<<nbN, T, 0, stream>>>(dinv, nNodes);

  // layer 1
  k_gemm1_wmma<<<gemmB, 128, 0, stream>>>(X, W1, h1, nNodes);
  k_self1<<<nbNH, T, 0, stream>>>(h1, dinv, agg1, nNodes);
  {
    long long work = (long long)E * 8;
    int nb = (int)((work + T - 1) / T);
    k_scatter1<<<nb, T, 0, stream>>>(dst, src, dinv, h1, agg1, E);
  }
  k_bias_relu<<<nbNH, T, 0, stream>>>(agg1, B1, nNodes);

  // layer 2
  k_gemm2_wmma<<<gemmB, 128, 0, stream>>>(agg1, W2, h2, nNodes);
  k_self2<<<nbNC, T, 0, stream>>>(h2, dinv, agg2, nNodes);
  {
    long long work = (long long)E * 4;
    int nb = (int)((work + T - 1) / T);
    k_scatter2<<<nb, T, 0, stream>>>(dst, src, dinv, h2, agg2, E);
  }
  k_logsoftmax<<<nbN, T, 0, stream>>>(agg2, B2, (float*)d_out, nNodes);
}